// MHFSpectralConv1D_39616778338530
// MI455X (gfx1250) — compile-verified
//
#include <hip/hip_runtime.h>

// MHFSpectralConv1D on gfx1250 (MI455X, wave32, WMMA).
//
// Only 32 rFFT modes survive, so fft/ifft collapse to two dense f32 GEMMs
// against a 64-column DFT basis (32 cos + 32 -sin), done with
// V_WMMA_F32_16X16X4_F32 to keep full fp32 precision over the K=8192
// reduction. HBM traffic is just x in (128MB) + out (128MB). The 2MB basis
// tables are L2-resident; each wave computes a 32-row M-pair so every B
// fragment is reused for two A fragments (halves L2 table traffic to 256MB
// per GEMM, balanced with the HBM stream).
//
// ws layout (floats): [T1p: 512K][T2: 512K][Xf: 256K][G: 256K] = 6 MB total.

#define LLEN   8192
#define NMODES 32
#define NF     64          // 32 real + 32 imag basis columns
#define NROWS  4096        // B * C = 64 * 64
#define KSPLIT 16          // stage-1 waves per block (K slices)
#define MPAIRS 128         // NROWS / 32
#define NCHUNK 32          // stage-3 split of L (256 cols per chunk)
#define TWO_PI 6.283185307179586476925f

#define T1P_OFF 0
#define T2_OFF  (LLEN * NF)
#define XF_OFF  (2 * LLEN * NF)
#define G_OFF   (2 * LLEN * NF + NROWS * NF)

typedef float v2f __attribute__((ext_vector_type(2)));
typedef float v8f __attribute__((ext_vector_type(8)));

// ---------------------------------------------------------------------------
// Basis tables.
// T2 [l*64 + j]        : j<32 -> cos(2*pi*j*l/L) ; j>=32 -> -sin(2*pi*(j-32)*l/L)
//   (l-major; stage-3 B frags load pairs (j, j+1) for fixed l as b64)
// T1p[(l>>1)*128 + j*2 + (l&1)] : same values, K-pair interleaved so stage-1
//   B frags load (l, l+1) for fixed j as a single b64.
// Phase reduced exactly: (j*l) mod 8192 in integer, then * (2*pi/8192).
// ---------------------------------------------------------------------------
__global__ __launch_bounds__(256) void k_tables(float* __restrict__ t1p,
                                                float* __restrict__ t2) {
  const int idx = blockIdx.x * 256 + threadIdx.x;   // over L*64
  const int l = idx >> 6;
  const int j = idx & 63;
  const int f = j & 31;
  const int r = (f * l) & (LLEN - 1);
  const float ph = (float)r * (TWO_PI / (float)LLEN);
  const float v = (j >= 32) ? -sinf(ph) : cosf(ph);
  t2[(size_t)l * NF + j] = v;
  t1p[(size_t)(l >> 1) * 128 + j * 2 + (l & 1)] = v;
}

// ---------------------------------------------------------------------------
// Stage 1: Xf[4096 x 64] = x[4096 x 8192] * D[8192 x 64] via f32 WMMA.
// One block per 32-row M-pair; 16 waves each own a K-slice of 512, then
// cross-wave reduce through LDS (128 KB).
// A frag (16x4):  lane&15 = M, v0/v1 = K+{0,1} (lanes>=16: K+{2,3}).
// B frag (4x16):  lane&15 = N, v0/v1 = K+{0,1} (lanes>=16: K+{2,3}).
// C/D  (16x16):   lane&15 = N, vgpr m = row m (lanes>=16: row m+8).
// ---------------------------------------------------------------------------
__global__ __launch_bounds__(512) void k_dft_fwd(const float* __restrict__ x,
                                                 const float* __restrict__ t1p,
                                                 float* __restrict__ xf) {
  __shared__ float red[KSPLIT * 32 * NF];           // 128 KB
  const int lane = threadIdx.x & 31;
  const int wid  = threadIdx.x >> 5;                // K slice id (0..15)
  const int hi   = lane >> 4;                       // 0 | 1
  const int ln   = lane & 15;
  const int rowbase = blockIdx.x * 32;
  const float* xrow0 = x + (size_t)(rowbase + ln) * LLEN;
  const float* xrow1 = x + (size_t)(rowbase + 16 + ln) * LLEN;

  v8f a0t0 = {}, a0t1 = {}, a0t2 = {}, a0t3 = {};   // M-tile 0 accumulators
  v8f a1t0 = {}, a1t1 = {}, a1t2 = {}, a1t3 = {};   // M-tile 1 accumulators
  const int k0beg = wid * (LLEN / KSPLIT);
  const int k0end = k0beg + (LLEN / KSPLIT);
  for (int k0 = k0beg; k0 < k0end; k0 += 4) {
    const int kb = k0 + 2 * hi;                     // even -> 8B aligned b64
    const v2f a0 = *(const v2f*)(xrow0 + kb);
    const v2f a1 = *(const v2f*)(xrow1 + kb);
    const float* tb = t1p + (size_t)(kb >> 1) * 128;
    const v2f b0 = *(const v2f*)(tb + (0 * 16 + ln) * 2);
    const v2f b1 = *(const v2f*)(tb + (1 * 16 + ln) * 2);
    const v2f b2 = *(const v2f*)(tb + (2 * 16 + ln) * 2);
    const v2f b3 = *(const v2f*)(tb + (3 * 16 + ln) * 2);
    a0t0 = __builtin_amdgcn_wmma_f32_16x16x4_f32(false, a0, false, b0, (short)0, a0t0, false, false);
    a0t1 = __builtin_amdgcn_wmma_f32_16x16x4_f32(false, a0, false, b1, (short)0, a0t1, false, false);
    a0t2 = __builtin_amdgcn_wmma_f32_16x16x4_f32(false, a0, false, b2, (short)0, a0t2, false, false);
    a0t3 = __builtin_amdgcn_wmma_f32_16x16x4_f32(false, a0, false, b3, (short)0, a0t3, false, false);
    a1t0 = __builtin_amdgcn_wmma_f32_16x16x4_f32(false, a1, false, b0, (short)0, a1t0, false, false);
    a1t1 = __builtin_amdgcn_wmma_f32_16x16x4_f32(false, a1, false, b1, (short)0, a1t1, false, false);
    a1t2 = __builtin_amdgcn_wmma_f32_16x16x4_f32(false, a1, false, b2, (short)0, a1t2, false, false);
    a1t3 = __builtin_amdgcn_wmma_f32_16x16x4_f32(false, a1, false, b3, (short)0, a1t3, false, false);
  }

  float* my = red + wid * (32 * NF);
  #pragma unroll
  for (int m = 0; m < 8; ++m) {
    const int r = m + 8 * hi;
    my[r * NF + 0 * 16 + ln] = a0t0[m];
    my[r * NF + 1 * 16 + ln] = a0t1[m];
    my[r * NF + 2 * 16 + ln] = a0t2[m];
    my[r * NF + 3 * 16 + ln] = a0t3[m];
    my[(16 + r) * NF + 0 * 16 + ln] = a1t0[m];
    my[(16 + r) * NF + 1 * 16 + ln] = a1t1[m];
    my[(16 + r) * NF + 2 * 16 + ln] = a1t2[m];
    my[(16 + r) * NF + 3 * 16 + ln] = a1t3[m];
  }
  __syncthreads();
  #pragma unroll
  for (int u = 0; u < 4; ++u) {
    const int idx = threadIdx.x + 512 * u;          // 2048 tile values
    float s = 0.0f;
    #pragma unroll
    for (int w = 0; w < KSPLIT; ++w) s += red[w * (32 * NF) + idx];
    xf[(size_t)blockIdx.x * (32 * NF) + idx] = s;
  }
}

// ---------------------------------------------------------------------------
// Stage 2: per-head complex mix, irfft scaling folded in.
// G[r, f]    = scale_f * Re(out_freq),  G[r, 32+f] = scale_f * Im(out_freq)
// scale_0 = 1/L, scale_f = 2/L.  (Im of bin 0 multiplies a zero basis column.)
// ---------------------------------------------------------------------------
__global__ __launch_bounds__(256) void k_mix(const float* __restrict__ xf,
                                             const float* __restrict__ wr,
                                             const float* __restrict__ wi,
                                             float* __restrict__ g) {
  const int idx = blockIdx.x * 256 + threadIdx.x;   // 64*4*16*32 = 131072
  const int f = idx & 31;
  const int o = (idx >> 5) & 15;
  const int h = (idx >> 9) & 3;
  const int b = idx >> 11;
  const float* xr0 = xf + (size_t)(b * 64 + h * 16) * NF + f;
  const float* wr0 = wr + ((size_t)(h * 16) * 16 + o) * NMODES + f;
  const float* wi0 = wi + ((size_t)(h * 16) * 16 + o) * NMODES + f;
  float or_ = 0.0f, oi_ = 0.0f;
  #pragma unroll
  for (int i = 0; i < 16; ++i) {
    const float xr  = xr0[i * NF];
    const float xi  = xr0[i * NF + 32];
    const float wre = wr0[i * (16 * NMODES)];
    const float wim = wi0[i * (16 * NMODES)];
    or_ += xr * wre - xi * wim;
    oi_ += xr * wim + xi * wre;
  }
  const float invL = 1.0f / (float)LLEN;
  const float scale = (f == 0) ? invL : 2.0f * invL;
  const size_t ro = (size_t)(b * 64 + h * 16 + o);
  g[ro * NF + f]      = or_ * scale;
  g[ro * NF + 32 + f] = oi_ * scale;
}

// ---------------------------------------------------------------------------
// Stage 3: out[4096 x 8192] = G[4096 x 64] * T2^T + bias, via f32 WMMA.
// Each wave owns (32-row M-pair, 256-column chunk); all 32 A frags of its
// G rows stay in registers; 16 N-tiles, each 2 independent chains of 16
// WMMAs sharing the B fragments.
// ---------------------------------------------------------------------------
__global__ __launch_bounds__(256) void k_idft(const float* __restrict__ g,
                                              const float* __restrict__ t2,
                                              const float* __restrict__ bias,
                                              float* __restrict__ out) {
  const int lane = threadIdx.x & 31;
  const int wid  = threadIdx.x >> 5;
  const int gw   = blockIdx.x * 8 + wid;
  const int mpair = gw >> 5;                        // / NCHUNK
  const int nc    = gw & (NCHUNK - 1);
  const int hi = lane >> 4, ln = lane & 15;
  const int rowbase = mpair * 32;

  v2f af0[16], af1[16];
  #pragma unroll
  for (int ks = 0; ks < 16; ++ks) {
    const int kb = ks * 4 + 2 * hi;
    af0[ks] = *(const v2f*)(g + (size_t)(rowbase + ln) * NF + kb);
    af1[ks] = *(const v2f*)(g + (size_t)(rowbase + 16 + ln) * NF + kb);
  }
  float bv0[8], bv1[8];
  #pragma unroll
  for (int m = 0; m < 8; ++m) {
    bv0[m] = bias[(rowbase + m + 8 * hi) & 63];
    bv1[m] = bias[(rowbase + 16 + m + 8 * hi) & 63];
  }

  const int colchunk = nc * (LLEN / NCHUNK);        // 256 columns
  for (int nt = 0; nt < 16; ++nt) {
    const int col = colchunk + nt * 16 + ln;
    const float* tcol = t2 + (size_t)col * NF;
    v8f acc0 = {}, acc1 = {};
    #pragma unroll
    for (int ks = 0; ks < 16; ++ks) {
      const int kb = ks * 4 + 2 * hi;
      const v2f bf = *(const v2f*)(tcol + kb);
      acc0 = __builtin_amdgcn_wmma_f32_16x16x4_f32(false, af0[ks], false, bf, (short)0, acc0, false, false);
      acc1 = __builtin_amdgcn_wmma_f32_16x16x4_f32(false, af1[ks], false, bf, (short)0, acc1, false, false);
    }
    #pragma unroll
    for (int m = 0; m < 8; ++m) {
      const int r = rowbase + m + 8 * hi;
      out[(size_t)r * LLEN + col]              = acc0[m] + bv0[m];
      out[(size_t)(r + 16) * LLEN + col]       = acc1[m] + bv1[m];
    }
  }
}

// ---------------------------------------------------------------------------
extern "C" void kernel_launch(void* const* d_in, const int* in_sizes, int n_in,
                              void* d_out, int out_size, void* d_ws, size_t ws_size,
                              hipStream_t stream) {
  const float* x    = (const float*)d_in[0];
  const float* wre  = (const float*)d_in[1];
  const float* wim  = (const float*)d_in[2];
  const float* bias = (const float*)d_in[3];
  float* out = (float*)d_out;
  float* ws  = (float*)d_ws;

  float* t1p = ws + T1P_OFF;
  float* t2  = ws + T2_OFF;
  float* xf  = ws + XF_OFF;
  float* g   = ws + G_OFF;

  k_tables <<<(LLEN * NF) / 256, 256, 0, stream>>>(t1p, t2);
  k_dft_fwd<<<MPAIRS, 512, 0, stream>>>(x, t1p, xf);
  k_mix    <<<(NROWS * NF / 2) / 256, 256, 0, stream>>>(xf, wre, wim, g);
  k_idft   <<<(MPAIRS * NCHUNK) / 8, 256, 0, stream>>>(g, t2, bias, out);
}